// Convolution_13365938225625
// MI455X (gfx1250) — compile-verified
//
#include <hip/hip_runtime.h>
#include <hip/hip_bf16.h>

#define GRID3D 96
#define MAPN (GRID3D * GRID3D * GRID3D)
#define C_IN 32
#define C_OUT 64

typedef float v2f __attribute__((ext_vector_type(2)));
typedef float v8f __attribute__((ext_vector_type(8)));

// ---------------------------------------------------------------------------
// Kernel 1: fill voxel->index map with -1
// ---------------------------------------------------------------------------
__global__ void init_map_kernel(int* __restrict__ map, int n) {
    int i = blockIdx.x * blockDim.x + threadIdx.x;
    if (i < n) map[i] = -1;
}

// ---------------------------------------------------------------------------
// Kernel 2: scatter input point indices into the dense map
// ---------------------------------------------------------------------------
__global__ void scatter_map_kernel(const int* __restrict__ pos,
                                   int* __restrict__ map, int n_in) {
    int i = blockIdx.x * blockDim.x + threadIdx.x;
    if (i < n_in) {
        int x = pos[3 * i + 0];
        int y = pos[3 * i + 1];
        int z = pos[3 * i + 2];
        map[(x * GRID3D + y) * GRID3D + z] = i;
    }
}

// ---------------------------------------------------------------------------
// Kernel 3: sparse conv via V_WMMA_F32_16X16X4_F32.
// Block = 256 threads = 8 waves. Each wave computes a 16-row x 64-col output
// tile; block covers 128 output rows. Per tap, the 32x64 f32 weight tile
// (8 KB) is staged into LDS cooperatively, then each wave runs
// 8 K-steps x 4 N-tiles of 16x16x4 f32 WMMA.
//
// A-fragment layout (16x4 f32, 2 VGPRs/lane):
//   lane<16 : M=lane,    K = 4k+{0,1}
//   lane>=16: M=lane-16, K = 4k+{2,3}
// B-fragment layout (4x16 f32, 2 VGPRs/lane):
//   lane<16 : N=lane,    K = 4k+{0,1}
//   lane>=16: N=lane-16, K = 4k+{2,3}
// C/D (16x16 f32, 8 VGPRs): vgpr r = rows {r, r+8}, N = lane&15.
// ---------------------------------------------------------------------------
__global__ __launch_bounds__(256) void sparse_conv_wmma_kernel(
    const float* __restrict__ feat,      // [n_in, 32]
    const float* __restrict__ outpos,    // [n_out, 3] (k + 0.5)
    const float* __restrict__ weight,    // [27, 32, 64]
    const int*  __restrict__ idx_map,    // [96^3]
    float* __restrict__ out,             // [n_out, 64]
    int n_out) {
    __shared__ float wlds[C_IN * C_OUT];  // 2048 floats = 8 KB

    const int tid  = threadIdx.x;
    const int lane = tid & 31;
    const int wv   = tid >> 5;      // wave id 0..7
    const int hi   = lane >> 4;     // 0 = K low pair, 1 = K high pair
    const int lrow = lane & 15;

    const int tile_base = blockIdx.x * 128 + wv * 16;
    const int m  = tile_base + lrow;
    const bool m_valid = (m < n_out);
    const int mc = m_valid ? m : (n_out - 1);   // clamped for safe loads

    const int ox = (int)outpos[mc * 3 + 0];
    const int oy = (int)outpos[mc * 3 + 1];
    const int oz = (int)outpos[mc * 3 + 2];

    v8f acc[4] = {};  // 4 N-tiles of 16 columns each

    for (int t = 0; t < 27; ++t) {
        // ---- stage this tap's 32x64 weight tile into LDS (512 float4) ----
        {
            const float4* wsrc = (const float4*)(weight + t * (C_IN * C_OUT));
            float4* wdst = (float4*)wlds;
            wdst[tid]       = wsrc[tid];
            wdst[tid + 256] = wsrc[tid + 256];
        }
        __syncthreads();

        // ---- neighbor lookup for this tap (select, not branch: EXEC stays full)
        const int dx = t / 9 - 1;
        const int dy = (t / 3) % 3 - 1;
        const int dz = t % 3 - 1;
        const int px = ox + dx, py = oy + dy, pz = oz + dz;
        bool valid = m_valid &
                     (px >= 0) & (px < GRID3D) &
                     (py >= 0) & (py < GRID3D) &
                     (pz >= 0) & (pz < GRID3D);
        const int flat = ((valid ? px : 0) * GRID3D + (valid ? py : 0)) * GRID3D
                         + (valid ? pz : 0);
        const int ii = idx_map[flat];
        valid = valid & (ii >= 0);
        const float scale = valid ? 1.0f : 0.0f;
        const float2* rowp =
            (const float2*)(feat + (size_t)(valid ? ii : 0) * C_IN);

        // ---- gather A fragments: 8 x float2 (global_load_b64 per lane) ----
        v2f afrag[8];
#pragma unroll
        for (int k = 0; k < 8; ++k) {
            float2 v = rowp[2 * k + hi];   // floats at 4k + 2*hi + {0,1}
            v2f a;
            a.x = v.x * scale;
            a.y = v.y * scale;
            afrag[k] = a;
        }

        // ---- 8 K-steps x 4 N-tiles of f32 WMMA, B read from LDS ----
#pragma unroll
        for (int k = 0; k < 8; ++k) {
            const float* bb = wlds + (4 * k + 2 * hi) * C_OUT + lrow;
#pragma unroll
            for (int n = 0; n < 4; ++n) {
                v2f b;
                b.x = bb[16 * n];          // K = 4k+2*hi,   N = 16n+lrow
                b.y = bb[C_OUT + 16 * n];  // K = 4k+2*hi+1, N = 16n+lrow
                acc[n] = __builtin_amdgcn_wmma_f32_16x16x4_f32(
                    /*neg_a=*/false, afrag[k],
                    /*neg_b=*/false, b,
                    /*c_mod=*/(short)0, acc[n],
                    /*reuse_a=*/false, /*reuse_b=*/false);
            }
        }
        __syncthreads();   // protect LDS before next tap's staging
    }

    // ---- store: vgpr r holds rows {tile_base+r (lanes<16), tile_base+r+8} ----
#pragma unroll
    for (int r = 0; r < 8; ++r) {
        const int row = tile_base + r + 8 * hi;
        if (row < n_out) {
            float* op = out + (size_t)row * C_OUT + lrow;
#pragma unroll
            for (int n = 0; n < 4; ++n) {
                op[16 * n] = acc[n][r];
            }
        }
    }
}

// ---------------------------------------------------------------------------
// Kernel 4: second output = out_positions * 0.5
// ---------------------------------------------------------------------------
__global__ void half_pos_kernel(const float* __restrict__ src,
                                float* __restrict__ dst, int n) {
    int i = blockIdx.x * blockDim.x + threadIdx.x;
    if (i < n) dst[i] = src[i] * 0.5f;
}

// ---------------------------------------------------------------------------
extern "C" void kernel_launch(void* const* d_in, const int* in_sizes, int n_in_args,
                              void* d_out, int out_size, void* d_ws, size_t ws_size,
                              hipStream_t stream) {
    (void)n_in_args; (void)out_size; (void)ws_size;

    const float* feat   = (const float*)d_in[0];  // [n_in, 32] f32
    const int*   inpos  = (const int*)  d_in[1];  // [n_in, 3] i32
    const float* outpos = (const float*)d_in[2];  // [n_out, 3] f32
    const float* weight = (const float*)d_in[3];  // [27, 32, 64] f32

    const int n_in  = in_sizes[1] / 3;
    const int n_out = in_sizes[2] / 3;

    int* idx_map = (int*)d_ws;                    // needs 96^3 * 4 = ~3.4 MB
    float* out = (float*)d_out;

    init_map_kernel<<<(MAPN + 255) / 256, 256, 0, stream>>>(idx_map, MAPN);
    scatter_map_kernel<<<(n_in + 255) / 256, 256, 0, stream>>>(inpos, idx_map, n_in);

    const int nblocks = (n_out + 127) / 128;
    sparse_conv_wmma_kernel<<<nblocks, 256, 0, stream>>>(
        feat, outpos, weight, idx_map, out, n_out);

    half_pos_kernel<<<(n_out * 3 + 255) / 256, 256, 0, stream>>>(
        outpos, out + (size_t)n_out * C_OUT, n_out * 3);
}